// EX_66992899883366
// MI455X (gfx1250) — compile-verified
//
#include <hip/hip_runtime.h>
#include <hip/hip_bf16.h>
#include <math.h>

// Flip to 0 if the async-to-LDS inline-asm path ever fails to assemble.
#define USE_ASYNC_LDS_COPY 1

// Native 128-bit vector so __builtin_nontemporal_load/store lower to
// global_load_b128 / global_store_b128 with TH_*_NT cache policy.
typedef unsigned int uv4 __attribute__((ext_vector_type(4)));

// ---------------------------------------------------------------------------
// Round an f32 value to bf16 bits with round-to-nearest-even.
// Safe for +/-inf (stays inf) and overflow-to-inf; caller handles NaN.
// ---------------------------------------------------------------------------
__device__ __forceinline__ unsigned bf16_rne_bits(float f) {
  unsigned b = __float_as_uint(f);
  return (b + 0x7FFFu + ((b >> 16) & 1u)) >> 16;
}

// ---------------------------------------------------------------------------
// Kernel A (<<<128, 512>>>): build the full 64Ki-entry bf16->bf16 function
// table into d_ws. Exactly one thread per table entry, no runtime divisions.
// Mirrors the reference bit-for-bit in structure: per-set-bit factors
// bf16(exp(+/-2^(k-134))) (LUT clamped: idx<7 -> exp(0)=1, idx>=141 ->
// exp(+/-128)), multiplied in f32 in bit order 7..0, RNE-rounded to bf16;
// NaN inputs -> 0xFFFF.
// ---------------------------------------------------------------------------
__global__ void __launch_bounds__(512) EXPLUT_build(unsigned short* __restrict__ tab) {
  __shared__ float L[2][263];  // [sign][idx], bf16-valued factors widened to f32
  const int tid = threadIdx.x;
#pragma unroll
  for (int t = 0; t < 2; ++t) {
    const int i = tid + t * 512;
    if (i < 526) {
      const int s = (i >= 263);
      const int k = s ? i - 263 : i;
      double ex;
      if (k < 7)        ex = 0.0;
      else if (k < 141) ex = ldexp(1.0, k - 134);   // 2^-127 .. 2^6
      else              ex = 128.0;                 // clamped tail
      const double v = exp(s ? -ex : ex);           // f64 exp like numpy
      const float  f = (float)v;                    // exp(128)->+inf, exp(-128)->0
      L[s][k] = __uint_as_float(bf16_rne_bits(f) << 16);  // bf16 value as f32
    }
  }
  __syncthreads();

  const int u = blockIdx.x * 512 + tid;             // exactly 0..65535
  const unsigned xm = (unsigned)u & 0x7FFFu;
  unsigned short r;
  if (xm > 0x7F80u) {
    r = (unsigned short)0xFFFFu;                    // NaN in -> all-ones bits
  } else {
    const int      s = u >> 15;
    const unsigned e = xm >> 7;                     // biased exponent 0..255
    const unsigned m = ((unsigned)u & 0x7Fu) | 0x80u;  // significand w/ implicit 1
    float p = 1.0f;
#pragma unroll
    for (int k = 7; k >= 0; --k) {                  // reference bit order 7..0
      const unsigned idx = ((m >> k) & 1u) ? (e + (unsigned)k) : 0u;  // L[0]=1
      p *= L[s][idx];
    }
    r = (unsigned short)bf16_rne_bits(p);
  }
  tab[u] = r;
}

// ---------------------------------------------------------------------------
// Kernel B (<<<512, 512, 128KB>>>): async-copy the 128 KB table from (L2-hot)
// global into LDS, then stream the input with non-temporal b128 accesses:
// one ds_load_u16 per bf16 element, 8 elements per lane-iteration.
// 128 KB LDS -> 2 blocks/WGP; 512 threads -> 32 waves resident per WGP.
// ---------------------------------------------------------------------------
__global__ void __launch_bounds__(512) EXPLUT_apply(
    const uv4* __restrict__ in, uv4* __restrict__ out,
    const uv4* __restrict__ tabg, long nvec,
    const unsigned short* __restrict__ xs, unsigned short* __restrict__ ys, long n) {
  extern __shared__ unsigned char smem[];           // 131072 bytes
  unsigned short* tab = (unsigned short*)smem;

#if USE_ASYNC_LDS_COPY
  {
    __attribute__((address_space(3))) unsigned char* l3 =
        (__attribute__((address_space(3))) unsigned char*)smem;
    for (int i = threadIdx.x; i < 8192; i += 512) {  // 8192 x 16B = 128 KB
      unsigned lds_off =
          (unsigned)(unsigned long long)(l3 + (unsigned long long)i * 16ull);
      unsigned long long gaddr = (unsigned long long)(const void*)(tabg + i);
      asm volatile("global_load_async_to_lds_b128 %0, %1, off"
                   :: "v"(lds_off), "v"(gaddr) : "memory");
    }
    asm volatile("s_wait_asynccnt 0" ::: "memory");
  }
#else
  {
    uv4* t4 = (uv4*)smem;
    for (int i = threadIdx.x; i < 8192; i += 512) t4[i] = tabg[i];
  }
#endif
  __syncthreads();

  const long gid     = (long)blockIdx.x * 512 + threadIdx.x;
  const long gstride = (long)gridDim.x * 512;
  for (long i = gid; i < nvec; i += gstride) {
    __builtin_prefetch((const void*)(in + i + gstride), 0, 0);  // global_prefetch_b8
    const uv4 v = __builtin_nontemporal_load(in + i);           // b128, TH_LOAD_NT
    uv4 r;
    r.x = (unsigned)tab[v.x & 0xFFFFu] | ((unsigned)tab[v.x >> 16] << 16);
    r.y = (unsigned)tab[v.y & 0xFFFFu] | ((unsigned)tab[v.y >> 16] << 16);
    r.z = (unsigned)tab[v.z & 0xFFFFu] | ((unsigned)tab[v.z >> 16] << 16);
    r.w = (unsigned)tab[v.w & 0xFFFFu] | ((unsigned)tab[v.w >> 16] << 16);
    __builtin_nontemporal_store(r, out + i);                    // b128, TH_STORE_NT
  }
  // Scalar tail (N not divisible by 8); no-op for the 4096x8192 case.
  for (long i = nvec * 8 + gid; i < n; i += gstride) ys[i] = tab[xs[i]];
}

// ---------------------------------------------------------------------------
// Host-side launcher (graph-capture safe: only kernel launches on `stream`).
// ---------------------------------------------------------------------------
extern "C" void kernel_launch(void* const* d_in, const int* in_sizes, int n_in,
                              void* d_out, int out_size, void* d_ws, size_t ws_size,
                              hipStream_t stream) {
  (void)n_in; (void)out_size; (void)ws_size;
  const unsigned short* x = (const unsigned short*)d_in[0];
  unsigned short*       y = (unsigned short*)d_out;
  unsigned short*     tab = (unsigned short*)d_ws;   // 131072 bytes of scratch
  const long N    = (long)in_sizes[0];
  const long nvec = N >> 3;                          // 128-bit vector = 8 bf16

  // Build the 64Ki-entry function table (deterministic, rebuilt every call).
  EXPLUT_build<<<128, 512, 0, stream>>>(tab);

  // Apply: 512 blocks x 512 threads, 128 KB dynamic LDS each.
  EXPLUT_apply<<<512, 512, 131072, stream>>>(
      (const uv4*)x, (uv4*)y, (const uv4*)tab, nvec, x, y, N);
}